// GraphTransformerEncoder_10411000725948
// MI455X (gfx1250) — compile-verified
//
#include <hip/hip_runtime.h>

typedef __attribute__((ext_vector_type(2))) float v2f;
typedef __attribute__((ext_vector_type(8))) float v8f;
typedef __attribute__((ext_vector_type(4))) int   v4i;

#define HIDDIM 128
#define NHEAD  4
#define DHEAD  32
#define LN_EPS 1e-5f

#if __has_builtin(__builtin_amdgcn_global_load_async_to_lds_b128) && \
    __has_builtin(__builtin_amdgcn_s_wait_asynccnt)
#define USE_ASYNC_LDS 1
#else
#define USE_ASYNC_LDS 0
#endif

#define GAS __attribute__((address_space(1)))   // global ("__device__")
#define LAS __attribute__((address_space(3)))   // LDS    ("__shared__")

// ---------------------------------------------------------------------------
// Y[M,128] = A[M,K] @ W[K,128] + bias   (M % 16 == 0, K % 16 == 0, K <= 128)
// One block = one 16-row tile; A-tile staged once into LDS (async on CDNA5),
// 8 waves each compute one 16x16 column tile with V_WMMA_F32_16X16X4_F32.
// f32 WMMA operand layouts (ISA 7.12.2):
//   A 16x4 : lane&15 = row M, VGPR j = K = 2*(lane>>4) + j
//   B 4x16 : lane&15 = col N, VGPR j = K = 2*(lane>>4) + j
//   C 16x16: lane&15 = col N, VGPR j = row M = j + 8*(lane>>4)
// ---------------------------------------------------------------------------
__global__ void __launch_bounds__(256) gemm_wmma_f32(
    const float* __restrict__ A, const float* __restrict__ W,
    const float* __restrict__ bias, float* __restrict__ Y, int K) {
  __shared__ float As[16 * HIDDIM];            // 8 KB max (K<=128)
  const int tid  = threadIdx.x;
  const int lane = tid & 31;
  const int wave = tid >> 5;                   // 0..7 -> column tile
  const int half = lane >> 4;                  // 0/1
  const int l16  = lane & 15;

  // cooperative stage of the 16 x K A-tile into LDS, 16B chunks
  const float* __restrict__ gA = A + (size_t)blockIdx.x * 16 * K;
  const int nchunk = (16 * K) >> 2;            // float4 chunks: 256 (K=64) / 512 (K=128)
  for (int c = tid; c < nchunk; c += 256) {
#if USE_ASYNC_LDS
    __builtin_amdgcn_global_load_async_to_lds_b128(
        (GAS v4i*)(gA + c * 4), (LAS v4i*)(As + c * 4), 0, 0);
#else
    ((float4*)As)[c] = ((const float4*)gA)[c];
#endif
  }
#if USE_ASYNC_LDS
  __builtin_amdgcn_s_wait_asynccnt(0);
#endif
  __syncthreads();

  const float* __restrict__ Arow = As + l16 * K;        // LDS
  const float* __restrict__ Bcol = W + wave * 16 + l16; // global, stride HIDDIM

  v8f c = {};
  for (int k = 0; k < K; k += 4) {
    v2f a, b;
    a.x = Arow[k + 2 * half + 0];
    a.y = Arow[k + 2 * half + 1];
    b.x = Bcol[(size_t)(k + 2 * half + 0) * HIDDIM];
    b.y = Bcol[(size_t)(k + 2 * half + 1) * HIDDIM];
    c = __builtin_amdgcn_wmma_f32_16x16x4_f32(
        /*neg_a=*/false, a, /*neg_b=*/false, b,
        /*c_mod=*/(short)0, c, /*reuse_a=*/false, /*reuse_b=*/false);
  }

  const int col = wave * 16 + l16;
  const float bv = bias[col];
  float* __restrict__ Yt = Y + (size_t)(blockIdx.x * 16) * HIDDIM + col;
#pragma unroll
  for (int j = 0; j < 8; ++j) {
    const int m = j + 8 * half;
    Yt[(size_t)m * HIDDIM] = c[j] + bv;
  }
}

// ---------------------------------------------------------------------------
// init kernels (harness poisons ws -> re-init every call)
// ---------------------------------------------------------------------------
__global__ void init_softmax_state(float* __restrict__ mx, float* __restrict__ den, int n) {
  int t = blockIdx.x * blockDim.x + threadIdx.x;
  if (t < n) { mx[t] = -__builtin_inff(); den[t] = 0.0f; }
}

__global__ void init_accum2h(const float* __restrict__ h, float* __restrict__ acc, int n) {
  int t = blockIdx.x * blockDim.x + threadIdx.x;
  if (t < n) acc[t] = 2.0f * h[t];   // identity residual applied twice in reference
}

__global__ void zero_buf(float* __restrict__ p, int n) {
  int t = blockIdx.x * blockDim.x + threadIdx.x;
  if (t < n) p[t] = 0.0f;
}

__device__ __forceinline__ void atomic_max_f32(float* addr, float val) {
  unsigned int* ua = (unsigned int*)addr;
  unsigned int old = *ua;
  while (__uint_as_float(old) < val) {
    unsigned int assumed = old;
    old = atomicCAS(ua, assumed, __float_as_uint(val));
    if (old == assumed) break;
  }
}

// ---------------------------------------------------------------------------
// logits[e,h] = sum_d attn[h,d] * leaky_relu(fs[src,h,d] + fd[dst,h,d], 0.2)
// fused with segment-max into mx[dst,h].  One wave32 per edge, lane = d.
// ---------------------------------------------------------------------------
__global__ void edge_logits_max(const float* __restrict__ fs, const float* __restrict__ fd,
                                const int* __restrict__ src, const int* __restrict__ dst,
                                const float* __restrict__ attn, float* __restrict__ logits,
                                float* __restrict__ mx, int E) {
  int e    = (int)((blockIdx.x * (size_t)blockDim.x + threadIdx.x) >> 5);
  int lane = threadIdx.x & 31;
  if (e >= E) return;
  const int s = src[e], d = dst[e];
  const float* __restrict__ fr = fs + (size_t)s * HIDDIM;
  const float* __restrict__ gr = fd + (size_t)d * HIDDIM;
#pragma unroll
  for (int h = 0; h < NHEAD; ++h) {
    float v = fr[h * DHEAD + lane] + gr[h * DHEAD + lane];
    v = (v > 0.0f) ? v : 0.2f * v;
    float p = v * attn[h * DHEAD + lane];
#pragma unroll
    for (int off = 16; off > 0; off >>= 1) p += __shfl_xor(p, off, 32);
    if (lane == 0) {
      logits[(size_t)e * NHEAD + h] = p;
      atomic_max_f32(&mx[d * NHEAD + h], p);
    }
  }
}

__global__ void edge_exp_den(float* __restrict__ logits, const int* __restrict__ dst,
                             const float* __restrict__ mx, float* __restrict__ den, int e4) {
  int t = blockIdx.x * blockDim.x + threadIdx.x;
  if (t >= e4) return;
  int e = t >> 2, h = t & 3;
  int d = dst[e];
  float ex = __expf(logits[t] - mx[d * NHEAD + h]);
  logits[t] = ex;
  atomicAdd(&den[d * NHEAD + h], ex);
}

// acc[dst, h, d] += (ex[e,h]/den[dst,h]) * fs[src, h, d]   (wave per edge)
__global__ void edge_scatter(const float* __restrict__ fs, const int* __restrict__ src,
                             const int* __restrict__ dst, const float* __restrict__ ex,
                             const float* __restrict__ den, float* __restrict__ acc, int E) {
  int e    = (int)((blockIdx.x * (size_t)blockDim.x + threadIdx.x) >> 5);
  int lane = threadIdx.x & 31;
  if (e >= E) return;
  const int s = src[e], d = dst[e];
  const float* __restrict__ fr = fs + (size_t)s * HIDDIM;
  float* __restrict__ ar = acc + (size_t)d * HIDDIM;
#pragma unroll
  for (int h = 0; h < NHEAD; ++h) {
    float alpha = ex[(size_t)e * NHEAD + h] / den[d * NHEAD + h];
    atomicAdd(&ar[h * DHEAD + lane], alpha * fr[h * DHEAD + lane]);
  }
}

// y = relu(layernorm(x) * g + b), one wave per node, 4 features per lane
__global__ void ln_relu(const float* __restrict__ x, const float* __restrict__ g,
                        const float* __restrict__ b, float* __restrict__ y, int Nn) {
  int n    = (int)((blockIdx.x * (size_t)blockDim.x + threadIdx.x) >> 5);
  int lane = threadIdx.x & 31;
  if (n >= Nn) return;
  const float* __restrict__ xr = x + (size_t)n * HIDDIM;
  float v0 = xr[lane], v1 = xr[lane + 32], v2 = xr[lane + 64], v3 = xr[lane + 96];
  float s = v0 + v1 + v2 + v3;
#pragma unroll
  for (int off = 16; off > 0; off >>= 1) s += __shfl_xor(s, off, 32);
  const float m = s * (1.0f / HIDDIM);
  float d0 = v0 - m, d1 = v1 - m, d2 = v2 - m, d3 = v3 - m;
  float q = d0 * d0 + d1 * d1 + d2 * d2 + d3 * d3;
#pragma unroll
  for (int off = 16; off > 0; off >>= 1) q += __shfl_xor(q, off, 32);
  const float r = rsqrtf(q * (1.0f / HIDDIM) + LN_EPS);
  float* __restrict__ yr = y + (size_t)n * HIDDIM;
  yr[lane]      = fmaxf(d0 * r * g[lane]      + b[lane],      0.0f);
  yr[lane + 32] = fmaxf(d1 * r * g[lane + 32] + b[lane + 32], 0.0f);
  yr[lane + 64] = fmaxf(d2 * r * g[lane + 64] + b[lane + 64], 0.0f);
  yr[lane + 96] = fmaxf(d3 * r * g[lane + 96] + b[lane + 96], 0.0f);
}

// partial column sums of final h into pooled[128]
__global__ void pool_partial(const float* __restrict__ hf, float* __restrict__ pooled, int Nn) {
  const int j = threadIdx.x;          // 128 threads
  int n0 = blockIdx.x * 256;
  int n1 = n0 + 256; if (n1 > Nn) n1 = Nn;
  float s = 0.0f;
  for (int n = n0; n < n1; ++n) s += hf[(size_t)n * HIDDIM + j];
  atomicAdd(&pooled[j], s);
}

// graph_emb = relu(concat(pooled/N, gf@W_g+b_g) @ W_f1 + b_f1) @ W_f2 + b_f2
__global__ void final_mlp(const float* __restrict__ pooled, const float* __restrict__ gf,
                          const float* __restrict__ W_g, const float* __restrict__ b_g,
                          const float* __restrict__ W_f1, const float* __restrict__ b_f1,
                          const float* __restrict__ W_f2, const float* __restrict__ b_f2,
                          float invN, float* __restrict__ out) {
  __shared__ float comb[2 * HIDDIM];
  __shared__ float hid[HIDDIM];
  const int j = threadIdx.x;          // 128 threads
  float gh = b_g[j];
  for (int k = 0; k < 32; ++k) gh += gf[k] * W_g[k * HIDDIM + j];
  comb[j]          = pooled[j] * invN;
  comb[HIDDIM + j] = gh;
  __syncthreads();
  float a = b_f1[j];
  for (int k = 0; k < 2 * HIDDIM; ++k) a += comb[k] * W_f1[k * HIDDIM + j];
  hid[j] = fmaxf(a, 0.0f);
  __syncthreads();
  float o = b_f2[j];
  for (int k = 0; k < HIDDIM; ++k) o += hid[k] * W_f2[k * HIDDIM + j];
  out[j] = o;
}

// ---------------------------------------------------------------------------
extern "C" void kernel_launch(void* const* d_in, const int* in_sizes, int n_in,
                              void* d_out, int out_size, void* d_ws, size_t ws_size,
                              hipStream_t stream) {
  const float* node_feats  = (const float*)d_in[0];
  const float* graph_feats = (const float*)d_in[1];
  const int*   src         = (const int*)d_in[2];
  const int*   dst         = (const int*)d_in[3];
  const float* W_in  = (const float*)d_in[4];
  const float* b_in  = (const float*)d_in[5];
  const float* W_g   = (const float*)d_in[6];
  const float* b_g   = (const float*)d_in[7];
  const float* W_src = (const float*)d_in[8];
  const float* b_src = (const float*)d_in[9];
  const float* W_dst = (const float*)d_in[10];
  const float* b_dst = (const float*)d_in[11];
  const float* attn  = (const float*)d_in[12];
  const float* ln_g  = (const float*)d_in[13];
  const float* ln_b  = (const float*)d_in[14];
  const float* W_f1  = (const float*)d_in[15];
  const float* b_f1  = (const float*)d_in[16];
  const float* W_f2  = (const float*)d_in[17];
  const float* b_f2  = (const float*)d_in[18];

  const int NODE_DIM = 64;
  const int Nn = in_sizes[0] / NODE_DIM;     // 50000
  const int E  = in_sizes[2];                // 1,650,000
  float* out = (float*)d_out;

  // scratch carve-out
  float* ws = (float*)d_ws;
  size_t o = 0;
  float* hbuf   = ws + o; o += (size_t)Nn * HIDDIM;
  float* fs     = ws + o; o += (size_t)Nn * HIDDIM;
  float* fd     = ws + o; o += (size_t)Nn * HIDDIM;
  float* accum  = ws + o; o += (size_t)Nn * HIDDIM;
  float* logits = ws + o; o += (size_t)E * NHEAD;
  float* mx     = ws + o; o += (size_t)Nn * NHEAD;
  float* den    = ws + o; o += (size_t)Nn * NHEAD;
  float* pooled = ws + o; o += HIDDIM;
  (void)ws_size; (void)n_in; (void)out_size;

  const int mtiles     = Nn / 16;                    // 3125 row tiles
  const int egrid_wave = (E + 7) / 8;                // wave-per-edge, 8 waves/block
  const int e4         = E * NHEAD;
  const int n4         = Nn * NHEAD;
  const int nh         = Nn * HIDDIM;

  // h = node_feats @ W_in + b_in
  gemm_wmma_f32<<<mtiles, 256, 0, stream>>>(node_feats, W_in, b_in, hbuf, NODE_DIM);

  for (int l = 0; l < 3; ++l) {
    gemm_wmma_f32<<<mtiles, 256, 0, stream>>>(hbuf, W_src + (size_t)l * HIDDIM * HIDDIM,
                                              b_src + l * HIDDIM, fs, HIDDIM);
    gemm_wmma_f32<<<mtiles, 256, 0, stream>>>(hbuf, W_dst + (size_t)l * HIDDIM * HIDDIM,
                                              b_dst + l * HIDDIM, fd, HIDDIM);
    init_softmax_state<<<(n4 + 255) / 256, 256, 0, stream>>>(mx, den, n4);
    edge_logits_max<<<egrid_wave, 256, 0, stream>>>(fs, fd, src, dst, attn + l * HIDDIM,
                                                    logits, mx, E);
    edge_exp_den<<<(e4 + 255) / 256, 256, 0, stream>>>(logits, dst, mx, den, e4);
    init_accum2h<<<(nh + 255) / 256, 256, 0, stream>>>(hbuf, accum, nh);
    edge_scatter<<<egrid_wave, 256, 0, stream>>>(fs, src, dst, logits, den, accum, E);
    float* dsth = (l < 2) ? hbuf : (out + HIDDIM);   // final layer -> output tail
    ln_relu<<<(Nn + 7) / 8, 256, 0, stream>>>(accum, ln_g + l * HIDDIM, ln_b + l * HIDDIM, dsth, Nn);
  }

  zero_buf<<<1, 128, 0, stream>>>(pooled, HIDDIM);
  pool_partial<<<(Nn + 255) / 256, 128, 0, stream>>>(out + HIDDIM, pooled, Nn);
  final_mlp<<<1, 128, 0, stream>>>(pooled, graph_feats, W_g, b_g, W_f1, b_f1, W_f2, b_f2,
                                   1.0f / (float)Nn, out);
}